// RoIAlign_24300924961316
// MI455X (gfx1250) — compile-verified
//
#include <hip/hip_runtime.h>

typedef __attribute__((ext_vector_type(4))) unsigned int v4u;
typedef __attribute__((ext_vector_type(8))) int          v8i;
typedef __attribute__((ext_vector_type(4))) int          v4i;

constexpr int   B = 2, C = 256, H = 200, W = 304, K = 1024;
constexpr int   PH = 7, PW = 7;
constexpr float SCALE = 0.25f;
constexpr int   WAVES = 4;                 // waves per block (wave32)
constexpr int   ROW_F  = 2 * C;            // floats per gathered row: (xlo, xlo+1) x C
constexpr int   CELL_F = 8 * ROW_F;        // floats per staged cell: 8 rows
constexpr unsigned TD0 = (unsigned)B * H * W * C;  // dim0 bound (elements) - never trips
constexpr unsigned TD1 = (unsigned)B * H * W;      // row count

__device__ __forceinline__ void axis_interp(float coord, int size, int& lo, int& hi,
                                            float& w0, float& w1) {
  const float valid = (coord >= -1.0f && coord <= (float)size) ? 1.0f : 0.0f;
  const float c   = fminf(fmaxf(coord, 0.0f), (float)(size - 1));
  const float lof = floorf(c);
  lo = (int)lof;
  hi = min(lo + 1, size - 1);
  const float fr = c - lof;
  w0 = (1.0f - fr) * valid;
  w1 = fr * valid;
}

// ---------------- Main RoIAlign: TDM gather-mode DMA into LDS, double buffered ----------
__global__ __launch_bounds__(WAVES * 32) void roialign_tdm_kernel(
    const float* __restrict__ nhwc, const float* __restrict__ rois,
    float* __restrict__ out) {
  __shared__ float stage[WAVES * 2 * CELL_F];   // 128 KB: 4 waves x 2 buffers x 16 KB

  const int k    = blockIdx.x;
  const int lane = threadIdx.x & 31;
  const int wave = threadIdx.x >> 5;

  const float rb = rois[k * 5 + 0];
  const float x1 = rois[k * 5 + 1] * SCALE;
  const float y1 = rois[k * 5 + 2] * SCALE;
  const float x2 = rois[k * 5 + 3] * SCALE;
  const float y2 = rois[k * 5 + 4] * SCALE;
  const int   b  = (int)rb;
  const float bw = fmaxf(x2 - x1, 1.0f) * (1.0f / PW);
  const float bh = fmaxf(y2 - y1, 1.0f) * (1.0f / PH);

  // Tensor DMA descriptor group 1 (constant for the launch).  ISA 8.4 bitfields:
  // data_size=4B, tile_dim0=ROW_F, tile_dim1=8 (valid 32-bit indices),
  // tensor_dim0 = total elements (column OOB check never fires; workspace is padded),
  // tensor_dim1 = #rows, tensor_dim0_stride = C.
  v8i g1;
  g1[0] = 0x00020000;                                         // data_size = 2 (4 bytes)
  g1[1] = (int)((TD0 & 0xFFFFu) << 16);
  g1[2] = (int)((TD0 >> 16) | ((TD1 & 0xFFFFu) << 16));
  g1[3] = (int)((TD1 >> 16) | ((unsigned)ROW_F << 16));
  g1[4] = 8;                                                  // tile_dim1 = #indices
  g1[5] = C;                                                  // tensor_dim0_stride
  g1[6] = 0;
  g1[7] = 0;

  const unsigned long long gaddr = (unsigned long long)(const void*)nhwc;
  const unsigned gw2 = (unsigned)gaddr;
  const unsigned gw3 = ((unsigned)(gaddr >> 32) & 0x01FFFFFFu) | (2u << 30); // type=2

  // LDS byte offset of this wave's staging area (flat-pointer low 32 bits == LDS offset).
  // ptrtoint escapes &stage into the TDM descriptor; asm-launder keeps it opaque so the
  // compiler cannot reason about (or undef-fold) the TDM-written LDS contents.
  const int  sfirst  = wave * 2 * CELL_F;                     // float index, this wave
  unsigned   ldsBase = (unsigned)(unsigned long long)(const void*)&stage[sfirst];
  asm volatile("" : "+v"(ldsBase)::"memory");

  auto issue = [&](int cell, int buf) {
    const int ph = cell / PW, pw = cell % PW;
    int rows[8];
#pragma unroll
    for (int s = 0; s < 4; ++s) {
      const int   sy = s >> 1, sx = s & 1;
      const float yc = y1 + ((float)(ph * 2 + sy) + 0.5f) * 0.5f * bh;
      const float xc = x1 + ((float)(pw * 2 + sx) + 0.5f) * 0.5f * bw;
      int ylo, yhi, xlo, xhi;
      float t0, t1;
      axis_interp(yc, H, ylo, yhi, t0, t1);
      axis_interp(xc, W, xlo, xhi, t0, t1);
      rows[2 * s]     = (b * H + ylo) * W + xlo;  // row covers (y, xlo..xlo+1, all c)
      rows[2 * s + 1] = (b * H + yhi) * W + xlo;
    }
    // Group 0: count=1, gather_mode=1, 32-bit indices, lds_addr, 57-bit global addr.
    v4u g0 = {0xC0000001u, ldsBase + (unsigned)(buf * CELL_F * 4), gw2, gw3};
    v4i g2 = {rows[0], rows[1], rows[2], rows[3]};
    v4i g3 = {rows[4], rows[5], rows[6], rows[7]};
    v8i g4 = {0, 0, 0, 0, 0, 0, 0, 0};            // VADDR4 group: unused, NULL per ISA
    __builtin_amdgcn_tensor_load_to_lds(g0, g1, g2, g3, g4, 0);
  };

  int buf = 0;
  issue(wave, 0);
  for (int cell = wave; cell < PH * PW; cell += WAVES) {
    const int nxt = cell + WAVES;
    if (nxt < PH * PW) {
      issue(nxt, buf ^ 1);                       // prefetch next cell into other buffer
      __builtin_amdgcn_s_wait_tensorcnt(1);      // oldest (current cell) DMA complete
    } else {
      __builtin_amdgcn_s_wait_tensorcnt(0);
    }
    asm volatile("" ::: "memory");               // fence: LDS now holds the DMA'd tile

    const int sbase = sfirst + buf * CELL_F;     // float index of current cell buffer
    const int ph = cell / PW, pw = cell % PW;
    float4 acc[2];
#pragma unroll
    for (int j = 0; j < 2; ++j) acc[j] = make_float4(0.f, 0.f, 0.f, 0.f);

#pragma unroll
    for (int s = 0; s < 4; ++s) {
      const int   sy = s >> 1, sx = s & 1;
      const float yc = y1 + ((float)(ph * 2 + sy) + 0.5f) * 0.5f * bh;
      const float xc = x1 + ((float)(pw * 2 + sx) + 0.5f) * 0.5f * bw;
      int ylo, yhi, xlo, xhi;
      float wy0, wy1, wx0, wx1;
      axis_interp(yc, H, ylo, yhi, wy0, wy1);
      axis_interp(xc, W, xlo, xhi, wx0, wx1);   // wx1==0 whenever xlo+1 was clamped
      const int rA = sbase + 2 * s * ROW_F;     // row ylo: [xlo:C][xlo+1:C]
      const int rB = rA + ROW_F;                // row yhi
#pragma unroll
      for (int j = 0; j < 2; ++j) {
        const int c0 = 4 * lane + 128 * j;      // ds_load_b128, conflict-free
        const float4 v00 = *(const float4*)&stage[rA + c0];
        const float4 v01 = *(const float4*)&stage[rA + C + c0];
        const float4 v10 = *(const float4*)&stage[rB + c0];
        const float4 v11 = *(const float4*)&stage[rB + C + c0];
        acc[j].x += wy0 * (wx0 * v00.x + wx1 * v01.x) + wy1 * (wx0 * v10.x + wx1 * v11.x);
        acc[j].y += wy0 * (wx0 * v00.y + wx1 * v01.y) + wy1 * (wx0 * v10.y + wx1 * v11.y);
        acc[j].z += wy0 * (wx0 * v00.z + wx1 * v01.z) + wy1 * (wx0 * v10.z + wx1 * v11.z);
        acc[j].w += wy0 * (wx0 * v00.w + wx1 * v01.w) + wy1 * (wx0 * v10.w + wx1 * v11.w);
      }
    }

    float* op = out + (size_t)k * C * (PH * PW) + cell;
#pragma unroll
    for (int j = 0; j < 2; ++j) {
      const int c0 = 4 * lane + 128 * j;
      op[(size_t)(c0 + 0) * (PH * PW)] = acc[j].x * 0.25f;
      op[(size_t)(c0 + 1) * (PH * PW)] = acc[j].y * 0.25f;
      op[(size_t)(c0 + 2) * (PH * PW)] = acc[j].z * 0.25f;
      op[(size_t)(c0 + 3) * (PH * PW)] = acc[j].w * 0.25f;
    }
    buf ^= 1;
  }
}

// ---------------- NCHW -> NHWC layout pass (HBM streaming, coalesced both sides) --------
__global__ __launch_bounds__(256) void nchw_to_nhwc_kernel(const float* __restrict__ f,
                                                           float* __restrict__ o) {
  __shared__ float tile[32][33];
  const int b  = blockIdx.z / H;
  const int y  = blockIdx.z % H;
  const int x0 = blockIdx.x * 32;
  const int c0 = blockIdx.y * 32;
  const int tx = threadIdx.x, ty = threadIdx.y;
#pragma unroll
  for (int i = 0; i < 32; i += 8) {
    const int c = c0 + ty + i, x = x0 + tx;
    if (x < W) tile[ty + i][tx] = f[(((size_t)b * C + c) * H + y) * W + x];
  }
  __syncthreads();
#pragma unroll
  for (int i = 0; i < 32; i += 8) {
    const int x = x0 + ty + i, c = c0 + tx;
    if (x < W) o[(((size_t)b * H + y) * W + x) * C + c] = tile[tx][ty + i];
  }
}

// ---------------- Fallback: direct NCHW gather (used only if workspace too small) -------
__global__ __launch_bounds__(256) void roialign_direct_kernel(
    const float* __restrict__ f, const float* __restrict__ rois,
    float* __restrict__ out) {
  const int idx  = blockIdx.x;
  const int k    = idx / (PH * PW);
  const int cell = idx % (PH * PW);
  const int ph = cell / PW, pw = cell % PW;
  const int c  = threadIdx.x;

  const float x1 = rois[k * 5 + 1] * SCALE;
  const float y1 = rois[k * 5 + 2] * SCALE;
  const float x2 = rois[k * 5 + 3] * SCALE;
  const float y2 = rois[k * 5 + 4] * SCALE;
  const int   b  = (int)rois[k * 5 + 0];
  const float bw = fmaxf(x2 - x1, 1.0f) * (1.0f / PW);
  const float bh = fmaxf(y2 - y1, 1.0f) * (1.0f / PH);

  const float* base = f + ((size_t)b * C + c) * ((size_t)H * W);
  float acc = 0.0f;
#pragma unroll
  for (int s = 0; s < 4; ++s) {
    const int   sy = s >> 1, sx = s & 1;
    const float yc = y1 + ((float)(ph * 2 + sy) + 0.5f) * 0.5f * bh;
    const float xc = x1 + ((float)(pw * 2 + sx) + 0.5f) * 0.5f * bw;
    int ylo, yhi, xlo, xhi;
    float wy0, wy1, wx0, wx1;
    axis_interp(yc, H, ylo, yhi, wy0, wy1);
    axis_interp(xc, W, xlo, xhi, wx0, wx1);
    acc += wy0 * (wx0 * base[ylo * W + xlo] + wx1 * base[ylo * W + xhi]) +
           wy1 * (wx0 * base[yhi * W + xlo] + wx1 * base[yhi * W + xhi]);
  }
  out[((size_t)k * C + c) * (PH * PW) + cell] = acc * 0.25f;
}

extern "C" void kernel_launch(void* const* d_in, const int* in_sizes, int n_in,
                              void* d_out, int out_size, void* d_ws, size_t ws_size,
                              hipStream_t stream) {
  const float* feat = (const float*)d_in[0];
  const float* rois = (const float*)d_in[1];
  float* out = (float*)d_out;

  const size_t nhwcBytes = (size_t)B * H * W * C * sizeof(float);
  if (ws_size >= nhwcBytes + 4096) {          // pad: x+1 overrun of last row (weight 0)
    float* nhwc = (float*)d_ws;
    dim3 tb(32, 8, 1);
    dim3 tg((W + 31) / 32, C / 32, B * H);
    nchw_to_nhwc_kernel<<<tg, tb, 0, stream>>>(feat, nhwc);
    roialign_tdm_kernel<<<K, WAVES * 32, 0, stream>>>(nhwc, rois, out);
  } else {
    roialign_direct_kernel<<<K * PH * PW, 256, 0, stream>>>(feat, rois, out);
  }
}